// s2v_DQN_65489661329987
// MI455X (gfx1250) — compile-verified
//
#include <hip/hip_runtime.h>
#include <hip/hip_bf16.h>
#include <math.h>

// ---------------------------------------------------------------------------
// Problem constants (from reference): B=16, S=512, P=64, T=4
// ---------------------------------------------------------------------------
#define NB    16
#define NS    512
#define NP    64
#define NROW  (NB * NS)      // 8192
#define TSTEPS 4

typedef __attribute__((ext_vector_type(2))) float v2f;
typedef __attribute__((ext_vector_type(8))) float v8f;

#if defined(__has_builtin)
#if __has_builtin(__builtin_amdgcn_wmma_f32_16x16x4_f32)
#define HAVE_WMMA_F32 1
#endif
#endif

// ---------------------------------------------------------------------------
// msg[b,s,p] = sum_j relu(d[b,s,j]*w4[p]+b4[p]);  comp1[b,s,p]=state*w1+b1
// One block per (b,s) row. 256 threads = 64 p-lanes x 4 j-chunks.
// ---------------------------------------------------------------------------
__global__ __launch_bounds__(256)
void k_msg_comp1(const float* __restrict__ state, const float* __restrict__ dist,
                 const int* __restrict__ didx,
                 const float* __restrict__ w1, const float* __restrict__ b1,
                 const float* __restrict__ w4, const float* __restrict__ b4,
                 float* __restrict__ msg, float* __restrict__ comp1)
{
    __shared__ float drow[NS];
    __shared__ float part[4][NP];
    const int row = blockIdx.x;          // 0..8191
    const int b   = row >> 9;
    const int s   = row & (NS - 1);
    const int tid = threadIdx.x;

    const float* dptr = dist + ((size_t)didx[b] * NS + s) * NS;
    drow[tid]       = dptr[tid];
    drow[tid + 256] = dptr[tid + 256];
    __syncthreads();

    const int p = tid & (NP - 1);
    const int c = tid >> 6;              // 0..3
    const float w  = w4[p];
    const float bb = b4[p];
    float acc = 0.f;
    const int j0 = c * 128;
#pragma unroll 8
    for (int j = j0; j < j0 + 128; ++j)
        acc += fmaxf(fmaf(drow[j], w, bb), 0.f);
    part[c][p] = acc;
    __syncthreads();

    if (tid < NP) {
        const float m = part[0][tid] + part[1][tid] + part[2][tid] + part[3][tid];
        msg[(size_t)row * NP + tid]   = m;
        comp1[(size_t)row * NP + tid] = fmaf(state[row], w1[tid], b1[tid]);
    }
}

// ---------------------------------------------------------------------------
// Y[row,n] = act( X[row,:] @ W[:,n] + bias[n] + (ADD ? add[row,n] : 0) )
// M=8192, K=64, N=64.  Block: 128 threads = 4 waves, 16 rows x 64 cols.
// Each wave owns one 16x16 N-tile and chains 16 x V_WMMA_F32_16X16X4_F32.
// ---------------------------------------------------------------------------
template <bool RELU, bool HAS_ADD>
__global__ __launch_bounds__(128)
void k_gemm64(const float* __restrict__ X, const float* __restrict__ W,
              const float* __restrict__ bias, const float* __restrict__ add,
              float* __restrict__ Y)
{
    __shared__ float Wl[64 * 65];        // padded: bank-conflict-free column reads
    __shared__ float Al[16 * 65];        // padded: bank-conflict-free row gathers
    const int tid     = threadIdx.x;
    const int wave    = tid >> 5;        // 0..3 -> N-tile
    const int lane    = tid & 31;
    const int rowbase = blockIdx.x * 16;

    for (int i = tid; i < 64 * 64; i += 128) {
        const int k = i >> 6, n = i & 63;
        Wl[k * 65 + n] = W[i];
    }
    for (int i = tid; i < 16 * 64; i += 128) {
        const int r = i >> 6, k = i & 63;
        Al[r * 65 + k] = X[(size_t)(rowbase + r) * NP + k];
    }
    __syncthreads();

    const int n    = wave * 16 + (lane & 15);
    const int half = lane >> 4;          // 0: rows i, 1: rows i+8 (C/D layout)
    const int ar   = lane & 15;          // A-matrix M = lane%16

    v8f c;
#pragma unroll
    for (int i = 0; i < 8; ++i) {
        float v = bias[n];
        if (HAS_ADD) v += add[(size_t)(rowbase + i + half * 8) * NP + n];
        c[i] = v;
    }

#if defined(HAVE_WMMA_F32)
#pragma unroll
    for (int kk = 0; kk < 16; ++kk) {
        const int k0 = kk * 4;
        v2f a, bm;
        // A 16x4 layout: lanes 0-15 -> K={k0,k0+1}; lanes 16-31 -> K={k0+2,k0+3}
        a[0] = Al[ar * 65 + k0 + 2 * half];
        a[1] = Al[ar * 65 + k0 + 2 * half + 1];
        // B 4x16 layout (row-striped, mirroring A): lanes 0-15 K={k0,k0+1}, 16-31 K={k0+2,k0+3}
        bm[0] = Wl[(k0 + 2 * half) * 65 + n];
        bm[1] = Wl[(k0 + 2 * half + 1) * 65 + n];
        c = __builtin_amdgcn_wmma_f32_16x16x4_f32(false, a, false, bm,
                                                  (short)0, c, false, false);
    }
#else
#pragma unroll
    for (int i = 0; i < 8; ++i) {
        const int r = i + half * 8;
        float acc = c[i];
        for (int k = 0; k < 64; ++k)
            acc = fmaf(Al[r * 65 + k], Wl[k * 65 + n], acc);
        c[i] = acc;
    }
#endif

#pragma unroll
    for (int i = 0; i < 8; ++i) {
        float v = c[i];
        if (RELU) v = fmaxf(v, 0.f);
        Y[(size_t)(rowbase + i + half * 8) * NP + n] = v;
    }
}

// ---------------------------------------------------------------------------
// agg[b,s,p] = (sum_s' emb[b,s',p]) - emb[b,s,p].  One block per b.
// ---------------------------------------------------------------------------
__global__ __launch_bounds__(256)
void k_colsum_agg(const float* __restrict__ emb, float* __restrict__ agg)
{
    __shared__ float part[4][NP];
    __shared__ float cs[NP];
    const int b   = blockIdx.x;
    const int tid = threadIdx.x;
    const int p   = tid & (NP - 1);
    const int c   = tid >> 6;
    const float* e = emb + (size_t)b * NS * NP;

    float acc = 0.f;
    const int s0 = c * 128;
    for (int s = s0; s < s0 + 128; ++s) acc += e[s * NP + p];
    part[c][p] = acc;
    __syncthreads();
    if (tid < NP) cs[tid] = part[0][tid] + part[1][tid] + part[2][tid] + part[3][tid];
    __syncthreads();

    float* a = agg + (size_t)b * NS * NP;
    for (int i = tid; i < NS * NP; i += 256)
        a[i] = cs[i & (NP - 1)] - e[i];
}

// ---------------------------------------------------------------------------
// Per-(b,p) mean / std (ddof=1); embn = (emb-mean)/std (0 where std==0).
// Also emits colsum_embn[b,p] = sum_s embn (feeds comp6).  One block per b.
// ---------------------------------------------------------------------------
__global__ __launch_bounds__(256)
void k_norm(const float* __restrict__ emb, float* __restrict__ embn,
            float* __restrict__ colsum_embn)
{
    __shared__ float ps1[4][NP];
    __shared__ float ps2[4][NP];
    __shared__ float meanS[NP];
    __shared__ float invS[NP];
    const int b   = blockIdx.x;
    const int tid = threadIdx.x;
    const int p   = tid & (NP - 1);
    const int c   = tid >> 6;
    const float* e = emb + (size_t)b * NS * NP;

    float s1 = 0.f, s2 = 0.f;
    const int r0 = c * 128;
    for (int s = r0; s < r0 + 128; ++s) {
        const float v = e[s * NP + p];
        s1 += v; s2 += v * v;
    }
    ps1[c][p] = s1; ps2[c][p] = s2;
    __syncthreads();
    if (tid < NP) {
        const float S1 = ps1[0][tid] + ps1[1][tid] + ps1[2][tid] + ps1[3][tid];
        const float S2 = ps2[0][tid] + ps2[1][tid] + ps2[2][tid] + ps2[3][tid];
        const float m  = S1 * (1.0f / (float)NS);
        float var = (S2 - (float)NS * m * m) * (1.0f / (float)(NS - 1));
        var = fmaxf(var, 0.f);
        const float sd = sqrtf(var);
        meanS[tid] = m;
        invS[tid]  = (sd != 0.f) ? (1.f / sd) : 0.f;   // std==0 -> embn = 0
    }
    __syncthreads();

    float* en = embn + (size_t)b * NS * NP;
    float csacc = 0.f;
    for (int s = r0; s < r0 + 128; ++s) {
        const float v = (e[s * NP + p] - meanS[p]) * invS[p];
        en[s * NP + p] = v;
        csacc += v;
    }
    ps1[c][p] = csacc;
    __syncthreads();
    if (tid < NP)
        colsum_embn[b * NP + tid] = ps1[0][tid] + ps1[1][tid] + ps1[2][tid] + ps1[3][tid];
}

// ---------------------------------------------------------------------------
// qbase[b] = b5 + sum_p relu(colsum_embn[b,:]@w6 + b6)[p] * w5[p]
// One block per b, 64 threads.
// ---------------------------------------------------------------------------
__global__ __launch_bounds__(64)
void k_comp6_qbase(const float* __restrict__ colsum_embn,
                   const float* __restrict__ w6, const float* __restrict__ b6,
                   const float* __restrict__ w5, const float* __restrict__ b5,
                   float* __restrict__ qbase)
{
    __shared__ float red[NP];
    const int b = blockIdx.x;
    const int p = threadIdx.x;
    const float* cs = colsum_embn + b * NP;
    float v = b6[p];
    for (int k = 0; k < NP; ++k) v = fmaf(cs[k], w6[k * NP + p], v);
    red[p] = fmaxf(v, 0.f) * w5[p];
    __syncthreads();
    for (int off = 32; off > 0; off >>= 1) {
        if (p < off) red[p] += red[p + off];
        __syncthreads();
    }
    if (p == 0) qbase[b] = red[0] + b5[0];
}

// ---------------------------------------------------------------------------
// q[row] = qbase[b] + sum_p relu(comp7[row,p]) * w5[64+p]
// ---------------------------------------------------------------------------
__global__ __launch_bounds__(256)
void k_qout(const float* __restrict__ comp7, const float* __restrict__ qbase,
            const float* __restrict__ w5, float* __restrict__ q)
{
    const int row = blockIdx.x * blockDim.x + threadIdx.x;
    if (row >= NROW) return;
    const float* c7 = comp7 + (size_t)row * NP;
    float acc = qbase[row >> 9];
#pragma unroll 8
    for (int p = 0; p < NP; ++p)
        acc = fmaf(fmaxf(c7[p], 0.f), w5[NP + p], acc);
    q[row] = acc;
}

__global__ void k_zero(float* __restrict__ p, int n)
{
    const int i = blockIdx.x * blockDim.x + threadIdx.x;
    if (i < n) p[i] = 0.f;
}

// ---------------------------------------------------------------------------
extern "C" void kernel_launch(void* const* d_in, const int* in_sizes, int n_in,
                              void* d_out, int out_size, void* d_ws, size_t ws_size,
                              hipStream_t stream)
{
    (void)in_sizes; (void)n_in; (void)out_size; (void)ws_size;
    const float* state = (const float*)d_in[0];
    const float* dist  = (const float*)d_in[1];
    const float* w1    = (const float*)d_in[2];
    const float* b1    = (const float*)d_in[3];
    const float* w2    = (const float*)d_in[4];
    const float* b2    = (const float*)d_in[5];
    const float* w3    = (const float*)d_in[6];
    const float* b3    = (const float*)d_in[7];
    const float* w4    = (const float*)d_in[8];
    const float* b4    = (const float*)d_in[9];
    const float* w5    = (const float*)d_in[10];
    const float* b5    = (const float*)d_in[11];
    const float* w6    = (const float*)d_in[12];
    const float* b6    = (const float*)d_in[13];
    const float* w7    = (const float*)d_in[14];
    const float* b7    = (const float*)d_in[15];
    const int*   didx  = (const int*)d_in[16];
    // d_in[17] is T; fixed to 4 by the reference's T_STEPS constant.

    const size_t TILE = (size_t)NROW * NP;   // 524288 floats
    float* ws     = (float*)d_ws;
    float* msg    = ws + 0 * TILE;
    float* comp1  = ws + 1 * TILE;
    float* base   = ws + 2 * TILE;           // comp1 + comp4
    float* emb    = ws + 3 * TILE;
    float* agg    = ws + 4 * TILE;
    float* embn   = ws + 5 * TILE;
    float* comp7  = ws + 6 * TILE;
    float* cse    = ws + 7 * TILE;           // colsum_embn [B*P]
    float* qbase  = cse + NB * NP;           // [B]
    float* q      = (float*)d_out;

    // 1) msg + comp1
    k_msg_comp1<<<NROW, 256, 0, stream>>>(state, dist, didx, w1, b1, w4, b4, msg, comp1);
    // 2) base = msg @ w3 + b3 + comp1   (== comp1 + comp4)
    k_gemm64<false, true><<<NROW / 16, 128, 0, stream>>>(msg, w3, b3, comp1, base);
    // 3) emb = 0
    k_zero<<<(int)((TILE + 255) / 256), 256, 0, stream>>>(emb, (int)TILE);
    // 4) T message-passing steps: emb = relu(agg @ w2 + b2 + base)
    for (int t = 0; t < TSTEPS; ++t) {
        k_colsum_agg<<<NB, 256, 0, stream>>>(emb, agg);
        k_gemm64<true, true><<<NROW / 16, 128, 0, stream>>>(agg, w2, b2, base, emb);
    }
    // 5) normalize per (b,p); emit colsum of embn
    k_norm<<<NB, 256, 0, stream>>>(emb, embn, cse);
    // 6) qbase[b] = b5 + relu(comp6) . w5[:64]
    k_comp6_qbase<<<NB, 64, 0, stream>>>(cse, w6, b6, w5, b5, qbase);
    // 7) comp7 = embn @ w7 + b7
    k_gemm64<false, false><<<NROW / 16, 128, 0, stream>>>(embn, w7, b7, nullptr, comp7);
    // 8) q = qbase + relu(comp7) . w5[64:]
    k_qout<<<NROW / 256, 256, 0, stream>>>(comp7, qbase, w5, q);
}